// S2Convolution_85744727097729
// MI455X (gfx1250) — compile-verified
//
#include <hip/hip_runtime.h>
#include <math.h>

// ---------------------------------------------------------------------------
// S2Convolution for MI455X (gfx1250, wave32, WMMA).
//
// Pipeline (all scratch in d_ws, float32 throughout):
//   A) table kernels: wig_s2[128x1024], wig_so3[64x43680], SH F[32x1024] (cplx),
//      DFT cos/sin tables C/S[64x64]
//   B) xf  : 128-pt DFT of x over alpha, only m in [-31,31]  (8,6,128,63) cplx
//   C) y   : y[s,i,o] = scaling * sum_g kernel[i,o,g] * F[g,s]      (cplx)
//   D) x_l : contract betas with wig_s2                              (cplx)
//   E) z_l : contract feature dim with conj(y)                       (cplx)
//   F) fused scatter + inverse 2D DFT per (z,o,t) image:
//        Fq[p,q]   = sum_l wso3[t,l,m,n] * z_l[l,m,n,zo]   (LDS, 64x64 cplx)
//        P = Fr*C - Fi*S ; Q = Fr*S + Fi*C                 (WMMA f32 16x16x4)
//        out = C^T*P - S^T*Q + bias                        (WMMA f32 16x16x4)
//      -> the only stage with meaningful FLOPs (~52 GFLOP), runs on the fp32
//         matrix pipe; avoids ever materializing the 0.5 GB frequency cube.
// ---------------------------------------------------------------------------

typedef __attribute__((ext_vector_type(2))) float v2f;
typedef __attribute__((ext_vector_type(8))) float v8f;

#define PI_F 3.14159265358979323846f

// ---- workspace layout (float offsets) -------------------------------------
constexpr size_t N_WS2  = (size_t)128 * 1024;        // wig_s2
constexpr size_t N_WSO3 = (size_t)64 * 43680;        // wig_so3
constexpr size_t N_F1   = (size_t)32 * 1024;         // F real
constexpr size_t N_CS   = (size_t)64 * 64;           // cos / sin table
constexpr size_t N_XF   = (size_t)8 * 6 * 128 * 63 * 2;
constexpr size_t N_Y    = (size_t)1024 * 192 * 2;
constexpr size_t N_XL   = (size_t)1024 * 48 * 2;
constexpr size_t N_ZL   = (size_t)43680 * 256 * 2;

constexpr size_t OFF_WS2  = 0;
constexpr size_t OFF_WSO3 = OFF_WS2 + N_WS2;
constexpr size_t OFF_FRE  = OFF_WSO3 + N_WSO3;
constexpr size_t OFF_FIM  = OFF_FRE + N_F1;
constexpr size_t OFF_CT   = OFF_FIM + N_F1;
constexpr size_t OFF_ST   = OFF_CT + N_CS;
constexpr size_t OFF_XF   = OFF_ST + N_CS;
constexpr size_t OFF_Y    = OFF_XF + N_XF;
constexpr size_t OFF_XL   = OFF_Y + N_Y;
constexpr size_t OFF_ZL   = OFF_XL + N_XL;

// ---- helpers ---------------------------------------------------------------
__device__ __forceinline__ int lofs(int l) { return l * (4 * l * l - 1) / 3; }

__device__ __forceinline__ int scol_to_l(int scol) {
  int l = (int)sqrtf((float)scol);
  while ((l + 1) * (l + 1) <= scol) ++l;
  while (l * l > scol) --l;
  return l;
}

__device__ __forceinline__ int pair_to_l(int pr) {
  int l = 0;
  while (l < 31 && lofs(l + 1) <= pr) ++l;
  return l;
}

// Wigner small-d element d^l_{m,n}(beta), log-gamma based k-sum.
__device__ float dev_wigner_d(int l, int mm, int nn, float beta) {
  float ch = cosf(0.5f * beta);
  float sh = sinf(0.5f * beta);
  float lch = logf(ch), lsh = logf(sh);
  int kmin = (nn - mm) > 0 ? (nn - mm) : 0;
  int kmax = (l - mm) < (l + nn) ? (l - mm) : (l + nn);
  float pref = 0.5f * (lgammaf((float)(l + mm + 1)) + lgammaf((float)(l - mm + 1)) +
                       lgammaf((float)(l + nn + 1)) + lgammaf((float)(l - nn + 1)));
  float acc = 0.f;
  for (int k = kmin; k <= kmax; ++k) {
    float logden = lgammaf((float)(l + nn - k + 1)) + lgammaf((float)(k + 1)) +
                   lgammaf((float)(l - mm - k + 1)) + lgammaf((float)(mm - nn + k + 1));
    int pc = 2 * l + nn - mm - 2 * k;
    int ps = mm - nn + 2 * k;
    float v = expf(pref - logden + (float)pc * lch + (float)ps * lsh);
    acc += ((mm - nn + k) & 1) ? -v : v;
  }
  return acc;
}

// ---- table kernels ---------------------------------------------------------
__global__ void k_cs(float* ct, float* st) {
  int idx = blockIdx.x * 256 + threadIdx.x;
  if (idx >= 64 * 64) return;
  int r = idx >> 6, c = idx & 63;
  float th = (2.0f * PI_F / 64.0f) * (float)((r * c) & 63);
  ct[idx] = cosf(th);
  st[idx] = sinf(th);
}

__global__ void k_wig_s2(float* ws2) {
  int idx = blockIdx.x * 256 + threadIdx.x;
  if (idx >= 128 * 1024) return;
  int tt = idx >> 10;
  int scol = idx & 1023;
  int l = scol_to_l(scol);
  int m = scol - l * l - l;
  float beta = ((float)tt + 0.5f) * (PI_F / 128.0f);
  // quadrature weight (b_in = 64)
  float j2 = (float)(2 * tt + 1);
  float s = 0.f;
  for (int k = 0; k < 64; ++k)
    s += sinf(j2 * (float)(2 * k + 1) * (PI_F / 256.0f)) / (float)(2 * k + 1);
  float w = (2.0f / 64.0f) * sinf(PI_F * j2 / 256.0f) * s;
  ws2[tt * 1024 + scol] = w * dev_wigner_d(l, m, 0, beta);
}

__global__ void k_wig_so3(float* wso3) {
  int idx = blockIdx.x * 256 + threadIdx.x;
  if (idx >= 64 * 43680) return;
  int t = idx / 43680;
  int pr = idx % 43680;
  int l = pair_to_l(pr);
  int r = pr - lofs(l);
  int tw = 2 * l + 1;
  int m = r / tw - l;
  int n = r % tw - l;
  float beta = ((float)t + 0.5f) * (PI_F / 64.0f);
  wso3[idx] = (float)(2 * l + 1) * dev_wigner_d(l, m, n, beta);
}

__global__ void k_sh(float* fre, float* fim) {
  int idx = blockIdx.x * 256 + threadIdx.x;
  if (idx >= 32 * 1024) return;
  int g = idx >> 10;
  int scol = idx & 1023;
  int l = scol_to_l(scol);
  int m = scol - l * l - l;
  int bi = g >> 3, ai = g & 7;                        // grid: beta-major
  float beta = (float)(bi + 1) * (PI_F / 8.0f) / 4.0f;
  float alpha = (float)ai * (2.0f * PI_F / 8.0f);
  float d = dev_wigner_d(l, m, 0, beta);
  float coef = sqrtf((float)(2 * l + 1) / (4.0f * PI_F));
  fre[g * 1024 + scol] = coef * d * cosf((float)m * alpha);
  fim[g * 1024 + scol] = coef * d * sinf((float)m * alpha);
}

// ---- stage B: xf = FFT_alpha(x), m in [-31,31] ------------------------------
__global__ void k_xf(const float* __restrict__ x, float* __restrict__ xf) {
  int idx = blockIdx.x * 256 + threadIdx.x;
  if (idx >= 8 * 6 * 128 * 63) return;
  int j = idx % 63;
  int rest = idx / 63;
  int tt = rest % 128;
  int zf = rest / 128;
  int m = j - 31;
  const float* xr = x + ((size_t)zf * 128 + tt) * 128;
  float re = 0.f, im = 0.f;
  for (int a = 0; a < 128; ++a) {
    int am = ((a * m) % 128 + 128) % 128;
    float th = -(2.0f * PI_F / 128.0f) * (float)am;   // numpy fft sign
    re += xr[a] * cosf(th);
    im += xr[a] * sinf(th);
  }
  float* o = xf + ((size_t)zf * 128 + tt) * 126 + 2 * j;
  o[0] = re;
  o[1] = im;
}

// ---- stage C: y[s,i,o] ------------------------------------------------------
__global__ void k_y(const float* __restrict__ kern, const float* __restrict__ fre,
                    const float* __restrict__ fim, float* __restrict__ y) {
  int idx = blockIdx.x * 256 + threadIdx.x;
  if (idx >= 1024 * 192) return;
  int s = idx / 192;
  int io = idx % 192;                                  // i*32+o
  float scal = rsqrtf(49152.0f);                       // 1/sqrt(ngrid*fi*b_out^4/b_in^2)
  float yr = 0.f, yi = 0.f;
  for (int g = 0; g < 32; ++g) {
    float kv = kern[io * 32 + g];
    yr += kv * fre[g * 1024 + s];
    yi += kv * fim[g * 1024 + s];
  }
  y[(size_t)(s * 192 + io) * 2 + 0] = yr * scal;
  y[(size_t)(s * 192 + io) * 2 + 1] = yi * scal;
}

// ---- stage D: x_l -----------------------------------------------------------
__global__ void k_xl(const float* __restrict__ ws2, const float* __restrict__ xf,
                     float* __restrict__ xl) {
  int idx = blockIdx.x * 256 + threadIdx.x;
  if (idx >= 1024 * 48) return;
  int scol = idx / 48;
  int zf = idx % 48;
  int l = scol_to_l(scol);
  int m = scol - l * l - l;
  int j = m + 31;
  float re = 0.f, im = 0.f;
  for (int tt = 0; tt < 128; ++tt) {
    float w = ws2[tt * 1024 + scol];
    const float* xp = xf + ((size_t)zf * 128 + tt) * 126 + 2 * j;
    re += w * xp[0];
    im += w * xp[1];
  }
  xl[(size_t)scol * 96 + zf * 2 + 0] = re;
  xl[(size_t)scol * 96 + zf * 2 + 1] = im;
}

// ---- stage E: z_l -----------------------------------------------------------
__global__ void k_zl(const float* __restrict__ xl, const float* __restrict__ y,
                     float* __restrict__ zl) {
  int fidx = blockIdx.x * 256 + threadIdx.x;
  if (fidx >= 43680 * 256) return;
  int pr = fidx >> 8;
  int zo = fidx & 255;
  int l = pair_to_l(pr);
  int r = pr - lofs(l);
  int tw = 2 * l + 1;
  int mi = r / tw;                                     // m+l
  int ni = r % tw;                                     // n+l
  int z = zo >> 5;
  int o = zo & 31;
  int sm = l * l + mi;
  int sn = l * l + ni;
  float zr = 0.f, zi = 0.f;
  for (int i = 0; i < 6; ++i) {
    float xr = xl[(size_t)sm * 96 + (z * 6 + i) * 2 + 0];
    float xi = xl[(size_t)sm * 96 + (z * 6 + i) * 2 + 1];
    float yr = y[(size_t)(sn * 192 + i * 32 + o) * 2 + 0];
    float yi = y[(size_t)(sn * 192 + i * 32 + o) * 2 + 1];
    zr += xr * yr + xi * yi;                           // x * conj(y)
    zi += xi * yr - xr * yi;
  }
  zl[(size_t)pr * 512 + zo * 2 + 0] = zr;
  zl[(size_t)pr * 512 + zo * 2 + 1] = zi;
}

// ---- stage F: fused scatter + inverse 2D DFT (WMMA f32 16x16x4) ------------
#define LSTR 65  // padded LDS stride (kills 64-bank conflicts on column walks)

__device__ __forceinline__ v8f vzero8() {
  v8f v = {0.f, 0.f, 0.f, 0.f, 0.f, 0.f, 0.f, 0.f};
  return v;
}

// acc += A(64xK tile rows i0..i0+15) * B(Kx64 tile cols j0..j0+15), K = 64.
// A fragment: lanes 0-15 K={k0,k0+1}, lanes 16-31 K={k0+2,k0+3} (ISA 16x4 f32).
// B fragment: row-striped like C/D (vgpr v: rows v / v+2).
__device__ __forceinline__ v8f wmma_gemm64(const float* A, int sA, const float* B,
                                           int sB, int i0, int j0, int lane, v8f acc) {
  const int row = i0 + (lane & 15);
  const int col = j0 + (lane & 15);
  const int hi2 = (lane >> 4) << 1;
#pragma unroll
  for (int k0 = 0; k0 < 64; k0 += 4) {
    v2f a, b;
    a.x = A[row * sA + k0 + hi2];
    a.y = A[row * sA + k0 + hi2 + 1];
    b.x = B[(k0 + hi2) * sB + col];
    b.y = B[(k0 + hi2 + 1) * sB + col];
    acc = __builtin_amdgcn_wmma_f32_16x16x4_f32(false, a, false, b, (short)0, acc,
                                                false, false);
  }
  return acc;
}

__global__ __launch_bounds__(256) void k_so3(const float* __restrict__ zl,
                                             const float* __restrict__ wso3,
                                             const float* __restrict__ ct,
                                             const float* __restrict__ st,
                                             const float* __restrict__ bias,
                                             float* __restrict__ out) {
  __shared__ float Fr[64 * LSTR];
  __shared__ float Fi[64 * LSTR];

  const int bid = blockIdx.x;          // bid = zo*64 + t
  const int t = bid & 63;
  const int zo = bid >> 6;
  const int o = zo & 31;

  // phase 0: zero the frequency tile
  for (int i = threadIdx.x; i < 64 * LSTR; i += 256) {
    Fr[i] = 0.f;
    Fi[i] = 0.f;
  }
  __syncthreads();

  // phase 1: Fq[p,q] = sum_l wso3[t,l,m,n] * z_l[l,m,n,zo]  (scatter, L2-bound)
  const float* wt = wso3 + (size_t)t * 43680;
  for (int pair = threadIdx.x; pair < 63 * 63; pair += 256) {
    int m = pair / 63 - 31;
    int n = pair % 63 - 31;
    int am = m < 0 ? -m : m;
    int an = n < 0 ? -n : n;
    int lmin = am > an ? am : an;
    float ar = 0.f, ai = 0.f;
    for (int l = lmin; l < 32; ++l) {
      int idx = lofs(l) + (m + l) * (2 * l + 1) + (n + l);
      const float* zp = zl + ((size_t)idx * 256 + zo) * 2;
      __builtin_prefetch(zp, 0, 0);
      float w = wt[idx];
      ar += w * zp[0];
      ai += w * zp[1];
    }
    int p = m & 63;
    int q = n & 63;
    Fr[p * LSTR + q] = ar;
    Fi[p * LSTR + q] = ai;
  }
  __syncthreads();

  const int lane = threadIdx.x & 31;
  const int wave = threadIdx.x >> 5;

  // phase 2a: P = Fr*C - Fi*S ; Q = Fr*S + Fi*C   (results held in registers)
  v8f pReg[2], qReg[2];
#pragma unroll
  for (int it = 0; it < 2; ++it) {
    int tile = wave + it * 8;
    int i0 = (tile >> 2) * 16;
    int j0 = (tile & 3) * 16;
    v8f p1 = wmma_gemm64(Fr, LSTR, ct, 64, i0, j0, lane, vzero8());
    v8f p2 = wmma_gemm64(Fi, LSTR, st, 64, i0, j0, lane, vzero8());
    v8f q1 = wmma_gemm64(Fr, LSTR, st, 64, i0, j0, lane, vzero8());
    v8f q2 = wmma_gemm64(Fi, LSTR, ct, 64, i0, j0, lane, vzero8());
    pReg[it] = p1 - p2;
    qReg[it] = q1 + q2;
  }
  __syncthreads();

  // phase 2b: park P,Q back into the (now dead) Fr/Fi LDS buffers
#pragma unroll
  for (int it = 0; it < 2; ++it) {
    int tile = wave + it * 8;
    int i0 = (tile >> 2) * 16;
    int j0 = (tile & 3) * 16;
    int col = j0 + (lane & 15);
    int hi = lane >> 4;
#pragma unroll
    for (int r = 0; r < 8; ++r) {
      Fr[(i0 + r + 8 * hi) * LSTR + col] = pReg[it][r];
      Fi[(i0 + r + 8 * hi) * LSTR + col] = qReg[it][r];
    }
  }
  __syncthreads();

  // phase 3: out[a,b] = sum_m C[a,m]*P[m,b] - S[a,m]*Q[m,b]  (+ bias)
  const float bo = bias[o];
  float* outp = out + (size_t)bid * 4096;
#pragma unroll
  for (int it = 0; it < 2; ++it) {
    int tile = wave + it * 8;
    int i0 = (tile >> 2) * 16;
    int j0 = (tile & 3) * 16;
    v8f r1 = wmma_gemm64(ct, 64, Fr, LSTR, i0, j0, lane, vzero8());  // C sym.
    v8f r2 = wmma_gemm64(st, 64, Fi, LSTR, i0, j0, lane, vzero8());  // S sym.
    v8f rr = r1 - r2;
    int col = j0 + (lane & 15);
    int hi = lane >> 4;
#pragma unroll
    for (int r = 0; r < 8; ++r)
      outp[(size_t)(i0 + r + 8 * hi) * 64 + col] = rr[r] + bo;
  }
}

// ---------------------------------------------------------------------------
extern "C" void kernel_launch(void* const* d_in, const int* in_sizes, int n_in,
                              void* d_out, int out_size, void* d_ws, size_t ws_size,
                              hipStream_t stream) {
  (void)in_sizes; (void)n_in; (void)out_size; (void)ws_size;
  const float* x = (const float*)d_in[0];
  const float* kern = (const float*)d_in[1];
  const float* bias = (const float*)d_in[2];
  float* out = (float*)d_out;
  float* ws = (float*)d_ws;

  float* ws2 = ws + OFF_WS2;
  float* wso3 = ws + OFF_WSO3;
  float* fre = ws + OFF_FRE;
  float* fim = ws + OFF_FIM;
  float* ct = ws + OFF_CT;
  float* st = ws + OFF_ST;
  float* xf = ws + OFF_XF;
  float* y = ws + OFF_Y;
  float* xl = ws + OFF_XL;
  float* zl = ws + OFF_ZL;

  dim3 blk(256);
  k_cs<<<16, blk, 0, stream>>>(ct, st);
  k_wig_s2<<<512, blk, 0, stream>>>(ws2);
  k_wig_so3<<<(64 * 43680 + 255) / 256, blk, 0, stream>>>(wso3);
  k_sh<<<128, blk, 0, stream>>>(fre, fim);
  k_xf<<<(8 * 6 * 128 * 63 + 255) / 256, blk, 0, stream>>>(x, xf);
  k_y<<<(1024 * 192 + 255) / 256, blk, 0, stream>>>(kern, fre, fim, y);
  k_xl<<<(1024 * 48 + 255) / 256, blk, 0, stream>>>(ws2, xf, xl);
  k_zl<<<(43680 * 256 + 255) / 256, blk, 0, stream>>>(xl, y, zl);
  k_so3<<<8 * 32 * 64, blk, 0, stream>>>(zl, wso3, ct, st, bias, out);
}